// PythonE80Cell_31138512896467
// MI455X (gfx1250) — compile-verified
//
#include <hip/hip_runtime.h>
#include <math.h>

typedef __attribute__((ext_vector_type(2))) float v2f;
typedef __attribute__((ext_vector_type(8))) float v8f;

#define T_STEPS 1024
#define BATCH   64
#define DIMK    512
#define NDIM    64
#define PROJ    (4 * NDIM)   // 256

// ---------------------------------------------------------------------------
// Kernel 1: fused projection GEMM, full f32 via V_WMMA_F32_16X16X4_F32.
// proj[(t*B+b)*256 + n] = sum_d x[(t*B+b)*512 + d] * W[n*512 + d]
// One wave computes one 16x16 output tile over K=512 (128 chained WMMAs).
// ---------------------------------------------------------------------------
__global__ __launch_bounds__(128)
void proj_gemm_wmma(const float* __restrict__ x, const float* __restrict__ W,
                    float* __restrict__ proj)
{
    const int lane   = threadIdx.x & 31;
    const int wave   = threadIdx.x >> 5;
    const int tile   = blockIdx.x * 4 + wave;   // 65536 tiles total
    const int tile_m = tile >> 4;               // 0..4095 (rows of 16)
    const int tile_n = tile & 15;               // 0..15   (cols of 16)

    const int l15  = lane & 15;
    const int koff = (lane >> 4) << 1;          // 0 or 2

    const float* arow = x + (size_t)(tile_m * 16 + l15) * DIMK;
    const float* brow = W + (size_t)(tile_n * 16 + l15) * DIMK;

    v8f acc = {};
#pragma unroll 8
    for (int k0 = 0; k0 < DIMK; k0 += 4) {
        v2f a = *reinterpret_cast<const v2f*>(arow + k0 + koff);
        v2f b = *reinterpret_cast<const v2f*>(brow + k0 + koff);
        acc = __builtin_amdgcn_wmma_f32_16x16x4_f32(false, a, false, b,
                                                    (short)0, acc, false, false);
    }

    // C/D layout: VGPR r -> M=r (lanes 0-15) / M=r+8 (lanes 16-31), N=lane&15
    const int n     = tile_n * 16 + l15;
    const int rbase = tile_m * 16 + ((lane >> 4) << 3);
#pragma unroll
    for (int r = 0; r < 8; ++r)
        proj[(size_t)(rbase + r) * PROJ + n] = acc[r];
}

// ---------------------------------------------------------------------------
// Kernel 2: recurrent scan, fully register-resident state.
// Block = half a batch: rows [half*32, half*32+32), 8 lanes per row, 8 j's per
// lane. S/M/B_S/B_M segments live in registers for all 1024 steps. All
// cross-thread reduction (norms + 5 matvecs) is wave-local shfl_xor over the
// 3 p-bits. One barrier per step (double-buffered LDS slice with register
// prefetch of step t+1 hiding global latency behind step t compute).
// ---------------------------------------------------------------------------
__device__ __forceinline__ float sigmoidf_(float v) {
    return 1.0f / (1.0f + __expf(-v));
}

__global__ __launch_bounds__(256)
void scan_kernel(const float* __restrict__ proj,
                 const float* __restrict__ S0, const float* __restrict__ M0,
                 const float* __restrict__ B_S, const float* __restrict__ B_M,
                 float* __restrict__ out,
                 float* __restrict__ S_out, float* __restrict__ M_out)
{
    __shared__ float raw[2][PROJ];              // double-buffered k|v|q|m slice

    const int b    = blockIdx.x >> 1;           // batch
    const int half = blockIdx.x & 1;            // row half
    const int tid  = threadIdx.x;
    const int i    = half * 32 + (tid >> 3);    // row 0..63
    const int p    = tid & 7;                   // j-segment (lane bits [2:0])
    const int j0   = p * 8;

    // ---- register-resident state + biases ----
    float ss[8], ms[8], bs[8], bm[8];
    const size_t st_base = (size_t)b * 4096 + (size_t)i * 64 + j0;
    const int    bi_base = i * 64 + j0;
#pragma unroll
    for (int jj = 0; jj < 8; ++jj) {
        ss[jj] = S0[st_base + jj];
        ms[jj] = M0[st_base + jj];
        bs[jj] = B_S[bi_base + jj];
        bm[jj] = B_M[bi_base + jj];
    }

    // stage first projection slice
    raw[0][tid] = proj[(size_t)b * PROJ + tid];
    __syncthreads();

    for (int t = 0; t < T_STEPS; ++t) {
        const int cur = t & 1, nxt = cur ^ 1;

        // ---- stage this step's segments from LDS into registers ----
        float rk[8], rq[8], rmv[8];
#pragma unroll
        for (int jj = 0; jj < 8; ++jj) {
            rk[jj]  = raw[cur][j0 + jj];        // k
            rq[jj]  = raw[cur][128 + j0 + jj];  // q
            rmv[jj] = raw[cur][192 + j0 + jj];  // m
        }
        const float v_i = raw[cur][64 + i];     // v (row scalar)

        // ---- prefetch next slice (wait lands just before the store below) --
        const int tn = (t + 1 < T_STEPS) ? (t + 1) : (T_STEPS - 1);
        const float pv = proj[((size_t)tn * BATCH + b) * PROJ + tid];

        // ---- L2 norms of k and m: segment partials + p-lane reduction ----
        float sk = 0.f, sm2 = 0.f;
#pragma unroll
        for (int jj = 0; jj < 8; ++jj) {
            sk  += rk[jj]  * rk[jj];
            sm2 += rmv[jj] * rmv[jj];
        }
        sk  += __shfl_xor(sk, 1);  sk  += __shfl_xor(sk, 2);  sk  += __shfl_xor(sk, 4);
        sm2 += __shfl_xor(sm2, 1); sm2 += __shfl_xor(sm2, 2); sm2 += __shfl_xor(sm2, 4);
        const float invk = 1.0f / (sqrtf(sk)  + 1e-6f);
        const float invm = 1.0f / (sqrtf(sm2) + 1e-6f);

        float kn[8], mn[8];
#pragma unroll
        for (int jj = 0; jj < 8; ++jj) {
            kn[jj] = rk[jj]  * invk;
            mn[jj] = rmv[jj] * invm;
        }

        // ---- phase 1: Mk = M@kn, Mm = M@mn, Sk = S@kn (old state) ----
        float amk = 0.f, amm = 0.f, ask = 0.f;
#pragma unroll
        for (int jj = 0; jj < 8; ++jj) {
            amk += ms[jj] * kn[jj];
            amm += ms[jj] * mn[jj];
            ask += ss[jj] * kn[jj];
        }
        amk += __shfl_xor(amk, 1); amk += __shfl_xor(amk, 2); amk += __shfl_xor(amk, 4);
        amm += __shfl_xor(amm, 1); amm += __shfl_xor(amm, 2); amm += __shfl_xor(amm, 4);
        ask += __shfl_xor(ask, 1); ask += __shfl_xor(ask, 2); ask += __shfl_xor(ask, 4);
        const float sdel = v_i - ask;           // s_delta[i]
        const float mdel = sdel - amm;          // m_delta[i] (uses old M)

        // ---- S update: S = sigmoid(M + Mk*kn + B_S) * S + sdel*kn ----
#pragma unroll
        for (int jj = 0; jj < 8; ++jj) {
            float g = sigmoidf_(ms[jj] + amk * kn[jj] + bs[jj]);
            ss[jj] = g * ss[jj] + sdel * kn[jj];
        }

        // ---- phase 2: Sm = S_new@mn, Sq = S_new@q ----
        float a_sm = 0.f, a_sq = 0.f;
#pragma unroll
        for (int jj = 0; jj < 8; ++jj) {
            a_sm += ss[jj] * mn[jj];
            a_sq += ss[jj] * rq[jj];
        }
        a_sm += __shfl_xor(a_sm, 1); a_sm += __shfl_xor(a_sm, 2); a_sm += __shfl_xor(a_sm, 4);
        a_sq += __shfl_xor(a_sq, 1); a_sq += __shfl_xor(a_sq, 2); a_sq += __shfl_xor(a_sq, 4);
        if (p == 0)                               // out = Sq^2 * sigmoid(Sq)
            out[((size_t)t * BATCH + b) * NDIM + i] = a_sq * a_sq * sigmoidf_(a_sq);

        // ---- M update: M = sigmoid(S_new + Sm*mn + B_M) * M + mdel*mn ----
#pragma unroll
        for (int jj = 0; jj < 8; ++jj) {
            float g = sigmoidf_(ss[jj] + a_sm * mn[jj] + bm[jj]);
            ms[jj] = g * ms[jj] + mdel * mn[jj];
        }

        // ---- publish next slice, single barrier per step ----
        raw[nxt][tid] = pv;
        __syncthreads();
    }

    // ---- final state writeback (fully coalesced, 32B per thread) ----
#pragma unroll
    for (int jj = 0; jj < 8; ++jj) {
        S_out[st_base + jj] = ss[jj];
        M_out[st_base + jj] = ms[jj];
    }
}

// ---------------------------------------------------------------------------
extern "C" void kernel_launch(void* const* d_in, const int* in_sizes, int n_in,
                              void* d_out, int out_size, void* d_ws, size_t ws_size,
                              hipStream_t stream)
{
    (void)in_sizes; (void)n_in; (void)out_size; (void)ws_size;

    const float* x   = (const float*)d_in[0];   // [1024,64,512]
    const float* S0  = (const float*)d_in[1];   // [64,64,64]
    const float* M0  = (const float*)d_in[2];   // [64,64,64]
    const float* W   = (const float*)d_in[3];   // [256,512]
    const float* B_S = (const float*)d_in[4];   // [64,64]
    const float* B_M = (const float*)d_in[5];   // [64,64]

    float* out   = (float*)d_out;                               // [1024,64,64]
    float* S_out = out   + (size_t)T_STEPS * BATCH * NDIM;      // [64,64,64]
    float* M_out = S_out + (size_t)BATCH * NDIM * NDIM;         // [64,64,64]

    float* proj = (float*)d_ws;   // [65536, 256] f32 = 64 MB scratch

    // 65536 tiles of 16x16, 4 waves (tiles) per 128-thread block
    proj_gemm_wmma<<<16384, 128, 0, stream>>>(x, W, proj);
    // 64 batches x 2 row-halves, 256 threads (8 lanes per row)
    scan_kernel<<<BATCH * 2, 256, 0, stream>>>(proj, S0, M0, B_S, B_M,
                                               out, S_out, M_out);
}